// DeathRxnLayer_16277926052096
// MI455X (gfx1250) — compile-verified
//
#include <hip/hip_runtime.h>
#include <cstdint>

// Reference shapes (fixed): B=8192, N = NV+NH = 128.
#define BDIM 8192
#define NDIM 128              // floats per row
#define ROWS_PER_WAVE 8       // rows (512B each) written per wave

typedef float float4v __attribute__((ext_vector_type(4)));

// ncovTE: each wave32 owns 8 consecutive 512B rows; lane = float4-within-row.
// Per iteration the wave emits one fully contiguous 512B row store burst.
__global__ void __launch_bounds__(256)
ncov_te_kernel(const float* __restrict__ mu,
               const float* __restrict__ ncov,
               const int*   __restrict__ isp_ptr,
               float*       __restrict__ outNcov) {
    const uint32_t isp = (uint32_t)(*isp_ptr);

    // 16 bytes of zeros in LDS: data source for the async bulk-zero stores.
    __shared__ float4v zpad;
    if (threadIdx.x == 0) zpad = (float4v){0.f, 0.f, 0.f, 0.f};
    __syncthreads();
    // Low 32 bits of the flat shared address are the LDS byte offset.
    const uint32_t zoff = (uint32_t)(uintptr_t)&zpad;

    const uint32_t t    = blockIdx.x * blockDim.x + threadIdx.x;
    const uint32_t lane = t & 31u;                 // float4 index within a row
    const uint32_t row0 = (t >> 5) * ROWS_PER_WAVE;

    const uint32_t ispLane   = isp >> 2;
    const uint32_t ispSub    = isp & 3u;
    const bool     isColLane = (lane == ispLane);

#pragma unroll
    for (uint32_t i = 0; i < ROWS_PER_WAVE; ++i) {
        const uint32_t row = row0 + i;             // b*128 + r
        const uint32_t r   = row & (NDIM - 1u);
        const uint32_t b   = row >> 7;

        float4v* dst = (float4v*)outNcov + (size_t)row * (NDIM / 4) + lane;
        const float* srcRow = ncov + (size_t)b * NDIM * NDIM + (size_t)isp * NDIM;

        if (r == isp) {
            // Special row: v = -ncov[b, isp, col0..col0+3], diagonal patched.
            float4v v = ((const float4v*)srcRow)[lane];
            v = -v;
            if (isColLane) {
                v[ispSub] = -2.0f * srcRow[isp] + mu[(size_t)b * NDIM + isp];
            }
            __builtin_nontemporal_store(v, dst);
        } else if (isColLane) {
            // Ordinary row, float4 containing column isp: single scatter value.
            float4v v = {0.f, 0.f, 0.f, 0.f};
            v[ispSub] = -srcRow[r];
            __builtin_nontemporal_store(v, dst);
        } else {
            // Pure zero: CDNA5 async LDS -> global store (ASYNCcnt path),
            // non-temporal so 512MB of zeros doesn't thrash the 192MB L2.
            asm volatile("global_store_async_from_lds_b128 %0, %1, off th:TH_STORE_NT"
                         :
                         : "v"(dst), "v"(zoff)
                         : "memory");
        }
    }

    // Drain outstanding async stores (S_ENDPGM would also wait; explicit here).
    asm volatile("s_wait_asynccnt 0" ::: "memory");
}

// muTE: B*N/4 float4s, zero except element isp of each batch row.
__global__ void __launch_bounds__(256)
mu_te_kernel(const float* __restrict__ mu,
             const int*   __restrict__ isp_ptr,
             float*       __restrict__ outMu) {
    const uint32_t isp  = (uint32_t)(*isp_ptr);
    const uint32_t t    = blockIdx.x * blockDim.x + threadIdx.x; // [0, B*N/4)
    const uint32_t lane = t & 31u;
    const uint32_t b    = t >> 5;

    float4v v = {0.f, 0.f, 0.f, 0.f};
    if (lane == (isp >> 2)) {
        v[isp & 3u] = -mu[(size_t)b * NDIM + isp];
    }
    __builtin_nontemporal_store(v, (float4v*)outMu + t);
}

extern "C" void kernel_launch(void* const* d_in, const int* in_sizes, int n_in,
                              void* d_out, int out_size, void* d_ws, size_t ws_size,
                              hipStream_t stream) {
    (void)in_sizes; (void)n_in; (void)out_size; (void)d_ws; (void)ws_size;

    const float* mu   = (const float*)d_in[0];
    const float* ncov = (const float*)d_in[1];
    const int*   isp  = (const int*)d_in[2];

    float* outMu   = (float*)d_out;                   // B*N floats
    float* outNcov = outMu + (size_t)BDIM * NDIM;     // B*N*N floats

    // ncovTE: B*N*N/4 float4s, 8 per thread -> 4,194,304 threads.
    {
        const uint32_t totalThreads =
            (uint32_t)BDIM * NDIM * NDIM / 4u / ROWS_PER_WAVE;
        const uint32_t blocks = totalThreads / 256u;   // 16,384
        ncov_te_kernel<<<blocks, 256, 0, stream>>>(mu, ncov, isp, outNcov);
    }
    // muTE: B*N/4 float4s = 262,144 threads.
    {
        const uint32_t total  = (uint32_t)BDIM * NDIM / 4u;
        const uint32_t blocks = total / 256u;          // 1,024
        mu_te_kernel<<<blocks, 256, 0, stream>>>(mu, isp, outMu);
    }
}